// sub_SlotAttention_73538430042530
// MI455X (gfx1250) — compile-verified
//
#include <hip/hip_runtime.h>
#include <math.h>

#define BB    16
#define NN    4096
#define DD    256
#define NSL   8
#define HH    256
#define NITER 3
#define QSCALE 0.0625f      /* 256^-0.5 */
#define EPSA   1e-8f
#define LNEPS  1e-5f

typedef float v2f __attribute__((ext_vector_type(2)));
typedef float v8f __attribute__((ext_vector_type(8)));

// d_out flat layout (floats): slots | pr | pf | sf | attns
#define OUT_SLOTS 0
#define OUT_PR    (BB*NSL*DD)                 // 32768
#define OUT_PF    (2*BB*NSL*DD)               // 65536
#define OUT_SF    (OUT_PF + BB*NN*DD)         // 16842752
#define OUT_ATTN  (OUT_SF + BB*NN*DD)         // 33619968

__device__ __forceinline__ float wred(float v) {
#pragma unroll
  for (int m = 16; m > 0; m >>= 1) v += __shfl_xor(v, m, 32);
  return v;
}

// ---------- Kernel A: LayerNorm k_inp and v_inp (one wave per row) ----------
__global__ void ln_kv(const float* __restrict__ kin, const float* __restrict__ vin,
                      const float* __restrict__ gk, const float* __restrict__ bk,
                      const float* __restrict__ gv, const float* __restrict__ bv,
                      float* __restrict__ kout, float* __restrict__ vout) {
  int wave = threadIdx.x >> 5, lane = threadIdx.x & 31;
  long row = (long)blockIdx.x * 8 + wave;               // 0 .. B*N-1
  for (int tsel = 0; tsel < 2; ++tsel) {
    const float* src = tsel ? vin : kin;
    float*       dst = tsel ? vout : kout;
    const float* g   = tsel ? gv : gk;
    const float* bb  = tsel ? bv : bk;
    const float* sr = src + row * DD;
    float x[8]; float s = 0.f, ss = 0.f;
#pragma unroll
    for (int j = 0; j < 8; ++j) { float v = sr[j*32 + lane]; x[j] = v; s += v; ss += v*v; }
    s = wred(s); ss = wred(ss);
    float m = s * (1.f/DD);
    float var = ss * (1.f/DD) - m*m;
    float rstd = rsqrtf(var + LNEPS);
    float* dr = dst + row * DD;
#pragma unroll
    for (int j = 0; j < 8; ++j) { int c = j*32 + lane; dr[c] = (x[j]-m)*rstd*g[c] + bb[c]; }
  }
}

// ---------- init slots = broadcast(slots_mu) ----------
__global__ void init_slots(const float* __restrict__ mu, float* __restrict__ slots) {
  int i = blockIdx.x * 256 + threadIdx.x;               // 0..32767
  slots[i] = mu[i & (NSL*DD - 1)];
}

// ---------- Kernel B: q = (LN(slots)*g_s+b_s) @ Wq^T + bq, pre-scaled ----------
__global__ void compute_q(const float* __restrict__ slots,
                          const float* __restrict__ gs, const float* __restrict__ bs,
                          const float* __restrict__ Wq, const float* __restrict__ bq,
                          float* __restrict__ qbuf) {
  __shared__ float sln[NSL][DD];
  int b = blockIdx.x;
  int wave = threadIdx.x >> 5, lane = threadIdx.x & 31;
  // wave w LayerNorms slot row w
  const float* srow = slots + (b*NSL + wave) * DD;
  float x[8]; float s = 0.f, ss = 0.f;
#pragma unroll
  for (int j = 0; j < 8; ++j) { float v = srow[j*32 + lane]; x[j] = v; s += v; ss += v*v; }
  s = wred(s); ss = wred(ss);
  float m = s * (1.f/DD);
  float rstd = rsqrtf(ss*(1.f/DD) - m*m + LNEPS);
#pragma unroll
  for (int j = 0; j < 8; ++j) { int c = j*32 + lane; sln[wave][c] = (x[j]-m)*rstd*gs[c] + bs[c]; }
  __syncthreads();
  int t = threadIdx.x;                                  // output column d
  float acc[NSL];
#pragma unroll
  for (int i = 0; i < NSL; ++i) acc[i] = 0.f;
  const float* wrow = Wq + (long)t * DD;
  for (int kk = 0; kk < DD; ++kk) {
    float w = wrow[kk];
#pragma unroll
    for (int i = 0; i < NSL; ++i) acc[i] += sln[i][kk] * w;
  }
  float bqa = bq[t];
#pragma unroll
  for (int i = 0; i < NSL; ++i) qbuf[(b*NSL + i)*DD + t] = (acc[i] + bqa) * QSCALE;
}

// ---------- Kernel C: dots via WMMA f32 16x16x4, softmax over slot axis ----------
// grid = B*8 workgroups of 256 threads (8 waves); each wg owns 512 tokens of one batch.
__global__ void attn_kernel(const float* __restrict__ qbuf, const float* __restrict__ kln,
                            float* __restrict__ attn_all, float* __restrict__ part, int it) {
  __shared__ float qlds[16 * DD];
  int b = blockIdx.x >> 3, wgb = blockIdx.x & 7;
  for (int i = threadIdx.x; i < 16*DD; i += 256) {
    int r = i >> 8;
    qlds[i] = (r < NSL) ? qbuf[(b*NSL + r)*DD + (i & 255)] : 0.f;  // zero-pad rows 8..15
  }
  __syncthreads();
  int wave = threadIdx.x >> 5, lane = threadIdx.x & 31;
  int lrow = lane & 15, koff = (lane >> 4) * 2;
  const float* aptr = &qlds[lrow * DD + koff];          // A frag: M=lrow, K=4s+koff+{0,1}
  float wsum[NSL];
#pragma unroll
  for (int r = 0; r < NSL; ++r) wsum[r] = 0.f;
  float* attn_out = attn_all + (long)(it*BB + b) * NSL * NN;

  for (int tile = wave; tile < 32; tile += 8) {
    int tb = wgb*512 + tile*16;
    const float* bptr = kln + ((long)b*NN + tb + lrow) * DD + koff;  // B frag: N=token
    v8f c = {};
#pragma unroll 8
    for (int s = 0; s < 64; ++s) {
      v2f a  = *(const v2f*)(aptr + s*4);
      v2f bw = *(const v2f*)(bptr + s*4);
      c = __builtin_amdgcn_wmma_f32_16x16x4_f32(false, a, false, bw, (short)0, c, false, false);
    }
    // per-lane softmax over the 8 slot logits (lanes 0..15 = tokens tb..tb+15)
    float mx = c[0];
#pragma unroll
    for (int r = 1; r < NSL; ++r) mx = fmaxf(mx, c[r]);
    float e[NSL]; float se = 0.f;
#pragma unroll
    for (int r = 0; r < NSL; ++r) { e[r] = __expf(c[r] - mx); se += e[r]; }
    float inv = 1.f / se;
    bool valid = lane < 16;
    int tok = tb + lane;
#pragma unroll
    for (int r = 0; r < NSL; ++r) {
      float a_ = e[r]*inv + EPSA;
      if (valid) { attn_out[(long)r*NN + tok] = a_; wsum[r] += a_; }
    }
  }
  // deterministic partial sums per wave (lanes >=16 contributed 0)
  int wi = wgb*8 + wave;
#pragma unroll
  for (int r = 0; r < NSL; ++r) {
    float t = wred(wsum[r]);
    if (lane == 0) part[(b*NSL + r)*64 + wi] = t;
  }
}

// ---------- Kernel C2: fixed-order reduce of partials -> 1/S ----------
__global__ void reduce_S(const float* __restrict__ part, float* __restrict__ Sinv) {
  int t = threadIdx.x;                                  // 0..127 == (b,slot)
  float s = 0.f;
  for (int i = 0; i < 64; ++i) s += part[t*64 + i];
  Sinv[t] = 1.f / s;
}

// ---------- Kernel D: updates = (attn @ v_ln) * (1/S) via WMMA ----------
// 32 blocks * 8 waves = 256 jobs = (b, d-tile of 16)
__global__ void updates_kernel(const float* __restrict__ attn_all, const float* __restrict__ vln,
                               const float* __restrict__ Sinv, float* __restrict__ upd, int it) {
  int wave = threadIdx.x >> 5, lane = threadIdx.x & 31;
  int job = blockIdx.x * 8 + wave;
  int b = job >> 4, d0 = (job & 15) << 4;
  int slot = lane & 15, koff = (lane >> 4) * 2;
  float amask = (slot < NSL) ? 1.f : 0.f;               // zero-pad A rows 8..15
  const float* arow = attn_all + ((long)(it*BB + b)*NSL + (slot & 7)) * NN + koff;
  const float* vcol = vln + (long)b*NN*DD + d0 + slot;  // N column = d0 + (lane&15)
  v8f c = {};
#pragma unroll 4
  for (int s = 0; s < 1024; ++s) {
    int r0 = s*4 + koff;
    v2f a = *(const v2f*)(arow + s*4);
    a.x *= amask; a.y *= amask;
    v2f bw; bw.x = vcol[(long)r0 * DD]; bw.y = vcol[(long)(r0+1) * DD];
    c = __builtin_amdgcn_wmma_f32_16x16x4_f32(false, a, false, bw, (short)0, c, false, false);
  }
  if (lane < 16) {
#pragma unroll
    for (int r = 0; r < NSL; ++r)
      upd[(b*NSL + r)*DD + d0 + lane] = c[r] * Sinv[b*NSL + r];
  }
}

// ---------- Kernel E: GRUCell + LN + MLP residual, one block per slot row ----------
__global__ void gru_ffn(const float* __restrict__ upd, float* __restrict__ slots,
                        const float* __restrict__ Wih, const float* __restrict__ bih,
                        const float* __restrict__ Whh, const float* __restrict__ bhh,
                        const float* __restrict__ gff, const float* __restrict__ bff,
                        const float* __restrict__ W1, const float* __restrict__ b1,
                        const float* __restrict__ W2, const float* __restrict__ b2) {
  __shared__ float xr[DD], hh[DD], pre[DD], hid[DD], red[8];
  int row = blockIdx.x, t = threadIdx.x;
  int wave = t >> 5, lane = t & 31;
  xr[t] = upd[(long)row*DD + t];
  hh[t] = slots[(long)row*DD + t];
  __syncthreads();
  float gir = bih[t], giz = bih[DD+t], gin = bih[2*DD+t];
  float ghr = bhh[t], ghz = bhh[DD+t], ghn = bhh[2*DD+t];
  const float* wir = Wih + (long)t*DD;
  const float* wiz = Wih + (long)(DD+t)*DD;
  const float* win = Wih + (long)(2*DD+t)*DD;
  const float* whr = Whh + (long)t*DD;
  const float* whz = Whh + (long)(DD+t)*DD;
  const float* whn = Whh + (long)(2*DD+t)*DD;
  for (int kk = 0; kk < DD; ++kk) {
    float xv = xr[kk], hv = hh[kk];
    gir += xv*wir[kk]; giz += xv*wiz[kk]; gin += xv*win[kk];
    ghr += hv*whr[kk]; ghz += hv*whz[kk]; ghn += hv*whn[kk];
  }
  float r = 1.f / (1.f + __expf(-(gir + ghr)));
  float z = 1.f / (1.f + __expf(-(giz + ghz)));
  float n = tanhf(gin + r*ghn);
  float sv = (1.f - z)*n + z*hh[t];
  // block LayerNorm of sv
  float ps = wred(sv);
  if (lane == 0) red[wave] = ps;
  __syncthreads();
  float mean = 0.f; for (int i = 0; i < 8; ++i) mean += red[i];
  mean *= (1.f/DD);
  __syncthreads();
  float dv = sv - mean;
  float ps2 = wred(dv*dv);
  if (lane == 0) red[wave] = ps2;
  __syncthreads();
  float var = 0.f; for (int i = 0; i < 8; ++i) var += red[i];
  var *= (1.f/DD);
  float rstd = rsqrtf(var + LNEPS);
  pre[t] = dv*rstd*gff[t] + bff[t];
  __syncthreads();
  float acc = b1[t];
  const float* w1r = W1 + (long)t*DD;
  for (int kk = 0; kk < DD; ++kk) acc += pre[kk]*w1r[kk];
  hid[t] = fmaxf(acc, 0.f);
  __syncthreads();
  float acc2 = b2[t];
  const float* w2r = W2 + (long)t*HH;
  for (int kk = 0; kk < HH; ++kk) acc2 += hid[kk]*w2r[kk];
  slots[(long)row*DD + t] = sv + acc2;
}

// ---------- Kernel F: pf/sf = attn_last-weighted sums of prompt/slot rows ----------
__global__ void final_proj(const float* __restrict__ prompts, const float* __restrict__ slots,
                           const float* __restrict__ attn_last,
                           float* __restrict__ pf, float* __restrict__ sf) {
  __shared__ float pl[NSL][DD], sl[NSL][DD], w[NSL][16];
  int b = blockIdx.x >> 8;                    // 256 blocks per batch
  int p0 = (blockIdx.x & 255) << 4;           // 16 tokens per block
  for (int i = threadIdx.x; i < NSL*DD; i += 256) {
    pl[i >> 8][i & 255] = prompts[i];
    sl[i >> 8][i & 255] = slots[b*NSL*DD + i];
  }
  if (threadIdx.x < 128) {
    int i = threadIdx.x >> 4, p = threadIdx.x & 15;
    w[i][p] = attn_last[((long)b*NSL + i)*NN + p0 + p];
  }
  __syncthreads();
  int d = threadIdx.x;
  for (int p = 0; p < 16; ++p) {
    float ap = 0.f, as = 0.f;
#pragma unroll
    for (int i = 0; i < NSL; ++i) { float ww = w[i][p]; ap += pl[i][d]*ww; as += sl[i][d]*ww; }
    long o = ((long)b*NN + p0 + p)*DD + d;
    pf[o] = ap; sf[o] = as;
  }
}

// ---------- Kernel G: write slots + broadcast prompts outputs ----------
__global__ void write_small(const float* __restrict__ slots, const float* __restrict__ prompts,
                            float* __restrict__ out) {
  int i = blockIdx.x * 256 + threadIdx.x;     // 0..32767
  out[OUT_SLOTS + i] = slots[i];
  out[OUT_PR + i] = prompts[i & (NSL*DD - 1)];
}

extern "C" void kernel_launch(void* const* d_in, const int* in_sizes, int n_in,
                              void* d_out, int out_size, void* d_ws, size_t ws_size,
                              hipStream_t stream) {
  const float* k_inp   = (const float*)d_in[2];
  const float* v_inp   = (const float*)d_in[3];
  const float* slots_mu= (const float*)d_in[4];
  const float* prompts = (const float*)d_in[5];
  const float* Wq = (const float*)d_in[6];
  const float* bq = (const float*)d_in[7];
  const float* W_ih = (const float*)d_in[8];
  const float* b_ih = (const float*)d_in[9];
  const float* W_hh = (const float*)d_in[10];
  const float* b_hh = (const float*)d_in[11];
  const float* W1 = (const float*)d_in[12];
  const float* b1 = (const float*)d_in[13];
  const float* W2 = (const float*)d_in[14];
  const float* b2 = (const float*)d_in[15];
  const float* g_k = (const float*)d_in[16];
  const float* b_k = (const float*)d_in[17];
  const float* g_v = (const float*)d_in[18];
  const float* b_v = (const float*)d_in[19];
  const float* g_s = (const float*)d_in[20];
  const float* b_s = (const float*)d_in[21];
  const float* g_ff = (const float*)d_in[22];
  const float* b_ff = (const float*)d_in[23];

  float* out = (float*)d_out;
  // k_ln / v_ln live in the pf/sf output regions until final_proj overwrites them
  float* kln = out + OUT_PF;
  float* vln = out + OUT_SF;
  float* attn_all = out + OUT_ATTN;

  float* ws    = (float*)d_ws;
  float* slots = ws;                 // 32768
  float* qbuf  = ws + 32768;         // 32768
  float* upd   = ws + 65536;         // 32768
  float* Sinv  = ws + 98304;         // 128
  float* part  = ws + 98432;         // 8192

  ln_kv<<<(BB*NN)/8, 256, 0, stream>>>(k_inp, v_inp, g_k, b_k, g_v, b_v, kln, vln);
  init_slots<<<128, 256, 0, stream>>>(slots_mu, slots);
  for (int it = 0; it < NITER; ++it) {
    compute_q<<<BB, 256, 0, stream>>>(slots, g_s, b_s, Wq, bq, qbuf);
    attn_kernel<<<BB*8, 256, 0, stream>>>(qbuf, kln, attn_all, part, it);
    reduce_S<<<1, 128, 0, stream>>>(part, Sinv);
    updates_kernel<<<32, 256, 0, stream>>>(attn_all, vln, Sinv, upd, it);
    gru_ffn<<<BB*NSL, 256, 0, stream>>>(upd, slots, W_ih, b_ih, W_hh, b_hh,
                                        g_ff, b_ff, W1, b1, W2, b2);
  }
  final_proj<<<BB*256, 256, 0, stream>>>(prompts, slots,
                                         attn_all + (long)2*BB*NSL*NN,
                                         out + OUT_PF, out + OUT_SF);
  write_small<<<128, 256, 0, stream>>>(slots, prompts, out);
}